// DynamicGraphQualification_12876311953413
// MI455X (gfx1250) — compile-verified
//
#include <hip/hip_runtime.h>
#include <math.h>

#define N 2048
#define D 256
#define BATCH 8
#define NTILES (N / 16)   // 128 column tiles of 16
#define EPSF 1e-5f
#define DELTAF 2.0f

typedef __attribute__((ext_vector_type(2))) float v2f;
typedef __attribute__((ext_vector_type(8))) float v8f;

__device__ __forceinline__ float sigmoidf_(float x) {
    return 1.0f / (1.0f + __expf(-x));
}

// ---------------------------------------------------------------------------
// Zero the per-batch {sum, sumsq} accumulators (16 floats).
// ---------------------------------------------------------------------------
__global__ void init_stats_kernel(float* stats) {
    if (threadIdx.x < 16) stats[threadIdx.x] = 0.0f;
}

// ---------------------------------------------------------------------------
// static_mask[i][j] = (dot(S_i, S_j) > 0) ? 1 : 0      (relu(x) > 0  <=>  x > 0)
// One workgroup (8 waves) per 16-row strip; each wave owns 16 column tiles.
// GEMM via V_WMMA_F32_16X16X4_F32 over K = 256.
// ---------------------------------------------------------------------------
__global__ __launch_bounds__(256) void static_mask_kernel(
    const float* __restrict__ S, unsigned char* __restrict__ mask) {
    __shared__ float As[16 * 260];   // padded stride: bank-conflict-free A reads

    const int rowblock = blockIdx.x;
    const int tid = threadIdx.x;

    for (int i = tid; i < 16 * D; i += 256) {
        int r = i >> 8, c = i & 255;
        As[r * 260 + c] = S[(size_t)(rowblock * 16 + r) * D + c];
    }
    __syncthreads();

    const int w = tid >> 5, lane = tid & 31;
    const int half = lane >> 4, l16 = lane & 15;
    const float* ap = &As[l16 * 260 + 2 * half];

    for (int t = w; t < NTILES; t += 8) {
        const int colbase = t * 16;
        v8f c = {};
        const float* bp = S + (size_t)(colbase + l16) * D + 2 * half;
#pragma unroll 8
        for (int k0 = 0; k0 < D; k0 += 4) {
            v2f a = *(const v2f*)(ap + k0);
            v2f b = *(const v2f*)(bp + k0);
            c = __builtin_amdgcn_wmma_f32_16x16x4_f32(
                false, a, false, b, (short)0, c, false, false);
        }
#pragma unroll
        for (int r = 0; r < 8; ++r) {
            int grow = rowblock * 16 + r + 8 * half;
            int gcol = colbase + l16;
            mask[(size_t)grow * N + gcol] = (c[r] > 0.0f) ? 1 : 0;
        }
    }
}

// ---------------------------------------------------------------------------
// Per (batch, 16-row strip):
//   phase A: graph = cur @ prev^T (WMMA fp32), masked-relu + 1e-10 -> d_out,
//            row sums + diagonal captured in LDS.
//   mid:     threshold[row] = sigmoid(cur_row . pool) * diag/rowsum.
//   phase B: pos = (x>=0) ? sigmoid(x) : 0 with x = mv/rowsum - thr,
//            rewrite d_out with pos, accumulate per-batch {sum, sumsq}.
// ---------------------------------------------------------------------------
__global__ __launch_bounds__(256) void graph_kernel(
    const float* __restrict__ cur, const float* __restrict__ prev,
    const unsigned char* __restrict__ mask, const float* __restrict__ pool,
    float* __restrict__ out, float* __restrict__ stats) {
    __shared__ float As[16 * 260];
    __shared__ float sRow[16];
    __shared__ float sDiag[16];
    __shared__ float sThr[16];
    __shared__ float sInv[16];
    __shared__ float sDot[256];
    __shared__ float sAcc[2];

    const int rowblock = blockIdx.x;
    const int batch = blockIdx.y;
    const int tid = threadIdx.x;

    if (tid < 16) { sRow[tid] = 0.0f; sDiag[tid] = 1e-10f; }
    if (tid < 2)  { sAcc[tid] = 0.0f; }

    const float* curb  = cur  + (size_t)batch * N * D;
    const float* prevb = prev + (size_t)batch * N * D;
    float* outb = out + (size_t)batch * N * N;

    for (int i = tid; i < 16 * D; i += 256) {
        int r = i >> 8, c = i & 255;
        As[r * 260 + c] = curb[(size_t)(rowblock * 16 + r) * D + c];
    }
    __syncthreads();

    const int w = tid >> 5, lane = tid & 31;
    const int half = lane >> 4, l16 = lane & 15;
    const float* ap = &As[l16 * 260 + 2 * half];

    float rp[8];
#pragma unroll
    for (int r = 0; r < 8; ++r) rp[r] = 0.0f;

    // ---- phase A: GEMM + mask/relu, store masked values into d_out ----
    for (int t = w; t < NTILES; t += 8) {
        const int colbase = t * 16;
        v8f c = {};
        const float* bp = prevb + (size_t)(colbase + l16) * D + 2 * half;
#pragma unroll 8
        for (int k0 = 0; k0 < D; k0 += 4) {
            v2f a = *(const v2f*)(ap + k0);
            v2f b = *(const v2f*)(bp + k0);
            c = __builtin_amdgcn_wmma_f32_16x16x4_f32(
                false, a, false, b, (short)0, c, false, false);
        }
#pragma unroll
        for (int r = 0; r < 8; ++r) {
            int lrow = r + 8 * half;
            int grow = rowblock * 16 + lrow;
            int gcol = colbase + l16;
            float g = c[r];
            unsigned char mk = mask[(size_t)grow * N + gcol];
            float mv = (mk && g > 0.0f) ? g : 0.0f;
            mv += 1e-10f;
            outb[(size_t)grow * N + gcol] = mv;
            rp[r] += mv;
            if (gcol == grow) sDiag[lrow] = mv;
        }
    }

    // reduce row partials across each 16-lane group -> per-strip row sums
#pragma unroll
    for (int r = 0; r < 8; ++r) {
        float v = rp[r];
        v += __shfl_xor(v, 1); v += __shfl_xor(v, 2);
        v += __shfl_xor(v, 4); v += __shfl_xor(v, 8);
        if (l16 == 0) atomicAdd(&sRow[r + 8 * half], v);
    }

    // ---- thresholds: dot(cur_row, pool), 16 threads x 16-wide segments ----
    {
        int r = tid >> 4, seg = tid & 15;
        float p = 0.0f;
#pragma unroll
        for (int k = 0; k < 16; ++k)
            p += As[r * 260 + seg * 16 + k] * pool[seg * 16 + k];
        sDot[tid] = p;
    }
    __syncthreads();
    if (tid < 16) {
        float dot = 0.0f;
#pragma unroll
        for (int s = 0; s < 16; ++s) dot += sDot[tid * 16 + s];
        float inv = 1.0f / sRow[tid];
        sInv[tid] = inv;
        sThr[tid] = sigmoidf_(dot) * sDiag[tid] * inv;
    }
    __syncthreads();

    // ---- phase B: pos-mask in place + per-batch statistics ----
    float s1 = 0.0f, s2 = 0.0f;
    for (int t = w; t < NTILES; t += 8) {
        const int colbase = t * 16;
#pragma unroll
        for (int r = 0; r < 8; ++r) {
            int lrow = r + 8 * half;
            int grow = rowblock * 16 + lrow;
            int gcol = colbase + l16;
            size_t idx = (size_t)grow * N + gcol;
            float mv = outb[idx];                    // same-lane store->load, ordered
            float x = mv * sInv[lrow] - sThr[lrow];
            float pos = (x >= 0.0f) ? sigmoidf_(x) : 0.0f;
            outb[idx] = pos;
            s1 += pos;
            s2 += pos * pos;
        }
    }
    s1 += __shfl_xor(s1, 1); s1 += __shfl_xor(s1, 2);
    s1 += __shfl_xor(s1, 4); s1 += __shfl_xor(s1, 8); s1 += __shfl_xor(s1, 16);
    s2 += __shfl_xor(s2, 1); s2 += __shfl_xor(s2, 2);
    s2 += __shfl_xor(s2, 4); s2 += __shfl_xor(s2, 8); s2 += __shfl_xor(s2, 16);
    if (lane == 0) { atomicAdd(&sAcc[0], s1); atomicAdd(&sAcc[1], s2); }
    __syncthreads();
    if (tid == 0) {
        atomicAdd(&stats[batch * 2 + 0], sAcc[0]);
        atomicAdd(&stats[batch * 2 + 1], sAcc[1]);
    }
}

// ---------------------------------------------------------------------------
// In-place: phi = exp(2*(gamma*x_norm + beta)) * (pos>0 ? pos : 1)
// (sigmoid(x) >= 0.5 whenever x >= 0, so pos==0 identifies the neg branch.)
// ---------------------------------------------------------------------------
__global__ __launch_bounds__(256) void finalize_kernel(
    float* __restrict__ out, const float* __restrict__ stats,
    const float* __restrict__ gamma, const float* __restrict__ beta) {
    const int batch = blockIdx.y;
    const float invNN = 1.0f / ((float)N * (float)N);
    const float mean = stats[batch * 2 + 0] * invNN;
    const float var  = stats[batch * 2 + 1] * invNN - mean * mean;
    const float rstd = rsqrtf(var + EPSF);
    const float g = gamma[0], b0 = beta[0];

    float* ob = out + (size_t)batch * N * N;
    size_t base = ((size_t)blockIdx.x * 256 + threadIdx.x) * 4;
    float4 v = *(float4*)(ob + base);
    float vv[4] = {v.x, v.y, v.z, v.w};
#pragma unroll
    for (int i = 0; i < 4; ++i) {
        float pos = vv[i];
        float xn = (pos - mean) * rstd;
        float tau = __expf((g * xn + b0) * DELTAF);
        vv[i] = tau * ((pos > 0.0f) ? pos : 1.0f);
    }
    *(float4*)(ob + base) = make_float4(vv[0], vv[1], vv[2], vv[3]);
}

// ---------------------------------------------------------------------------
extern "C" void kernel_launch(void* const* d_in, const int* in_sizes, int n_in,
                              void* d_out, int out_size, void* d_ws, size_t ws_size,
                              hipStream_t stream) {
    (void)in_sizes; (void)n_in; (void)out_size; (void)ws_size;
    const float* S     = (const float*)d_in[0];   // [n, d]
    const float* cur   = (const float*)d_in[1];   // [b, n, d]
    const float* prev  = (const float*)d_in[2];   // [b, n, d]
    const float* pool  = (const float*)d_in[3];   // [d, 1]
    const float* gamma = (const float*)d_in[4];   // [1]
    const float* beta  = (const float*)d_in[5];   // [1]
    float* out = (float*)d_out;                   // [b, n, n]

    unsigned char* mask = (unsigned char*)d_ws;                    // n*n bytes = 4 MB
    float* stats = (float*)((char*)d_ws + (size_t)N * N);          // 16 floats

    init_stats_kernel<<<1, 32, 0, stream>>>(stats);
    static_mask_kernel<<<dim3(NTILES), 256, 0, stream>>>(S, mask);
    graph_kernel<<<dim3(NTILES, BATCH), 256, 0, stream>>>(cur, prev, mask, pool, out, stats);
    finalize_kernel<<<dim3(N * N / 1024, BATCH), 256, 0, stream>>>(out, stats, gamma, beta);
}